// A3C_DND_LSTM_75737453298419
// MI455X (gfx1250) — compile-verified
//
#include <hip/hip_runtime.h>
#include <hip/hip_bf16.h>
#include <math.h>

// ---------------- problem constants ----------------
#define H      512
#define KDIM   64
#define NACT   3
#define NMEM   100000
#define XDIM   (KDIM + NACT + 1)   // 68
#define NGATE  (5 * H)             // 2560

#define NTILES       (NMEM / 16)          // 6250 tiles of 16 keys
#define SCORE_BLOCKS ((NTILES + 7) / 8)   // 782 blocks, 8 waves each
#define RBLK         256                  // row-chunk blocks for vals pass
#define CHUNK        ((NMEM + RBLK - 1) / RBLK)   // 391 rows per block

// ---------------- workspace layout (float offsets) ----------------
#define OFF_FEATS 0                       // 64
#define OFF_XT    64                      // 68
#define OFF_SMAX  132                     // 1
#define OFF_MT    144                     // 512
#define OFF_GATES 656                     // 2560
#define OFF_S     3216                    // 100000 scores
#define OFF_MAXP  103216                  // 1024 (782 used)
#define OFF_SUMP  104240                  // 256
#define OFF_MPART 104496                  // 256*512 = 131072
// total ~235568 floats ~= 942 KB

typedef __attribute__((ext_vector_type(2))) float v2f;
typedef __attribute__((ext_vector_type(8))) float v8f;

#if defined(__gfx1250__) && __has_builtin(__builtin_amdgcn_wmma_f32_16x16x4_f32)
#define USE_WMMA_F32 1
#endif

__device__ __forceinline__ float sigmoidf_(float x) {
    return 1.0f / (1.0f + __expf(-x));
}

// ---------------- K0: MLP encoder + x_t assembly ----------------
__global__ void encoder_kernel(const float* __restrict__ obs,
                               const float* __restrict__ pa,
                               const float* __restrict__ pr,
                               const float* __restrict__ W1, const float* __restrict__ b1,
                               const float* __restrict__ W2, const float* __restrict__ b2,
                               float* __restrict__ feats_ws,
                               float* __restrict__ x_ws,
                               float* __restrict__ out) {
    __shared__ float hid[32];
    const int t = threadIdx.x;   // 64 threads
    if (t < 32) {
        float a = b1[t];
#pragma unroll
        for (int k = 0; k < 9; ++k) a = fmaf(W1[t * 9 + k], obs[k], a);
        hid[t] = fmaxf(a, 0.0f);
    }
    __syncthreads();
    {
        float a = b2[t];
#pragma unroll
        for (int k = 0; k < 32; ++k) a = fmaf(W2[t * 32 + k], hid[k], a);
        float f = fmaxf(a, 0.0f);
        feats_ws[t]   = f;
        x_ws[t]       = f;        // x_t[0:64] = feats
        out[1028 + t] = f;        // feats output slice
    }
    if (t < NACT) x_ws[KDIM + t] = pa[t];
    if (t == 0)   x_ws[KDIM + NACT] = pr[0];
}

// ---------------- K1: scores s_j = 2*k_j.f - ||k_j||^2 via WMMA f32 ----------------
// One wave handles a tile of 16 keys. B = 4x16 keys chunk (keys along N),
// A = 4-elem feats chunk broadcast over all 16 M rows -> D rows identical,
// so acc[0] in lane l is the dot product for key tile*16 + (l&15).
__global__ void score_kernel(const float* __restrict__ keys,
                             const float* __restrict__ feats,
                             float* __restrict__ sarr,
                             float* __restrict__ maxpart) {
    const int lane = threadIdx.x & 31;
    const int wave = threadIdx.x >> 5;
    const int tile = blockIdx.x * 8 + wave;
    __shared__ float red[8];

    float s = -INFINITY;
    if (tile < NTILES) {
        const int n = lane & 15;      // key within tile
        const int g = lane >> 4;      // K-half group
        const float* kp = keys + (size_t)(tile * 16 + n) * KDIM;

        float nrm = 0.0f;
#ifdef USE_WMMA_F32
        v8f acc = {};
#else
        float dotp = 0.0f;
#endif
#pragma unroll
        for (int cb = 0; cb < KDIM; cb += 4) {
            // A (16x4 f32): lane(m=*,g): vgpr j -> A[m][cb+2g+j]  (broadcast over m)
            // B (4x16 f32): lane(n,g):   vgpr j -> B[cb+2g+j][n] = keys[n][cb+2g+j]
            const float2 bq = *(const float2*)(kp + cb + 2 * g);
            const float2 aq = *(const float2*)(feats + cb + 2 * g);
            nrm = fmaf(bq.x, bq.x, nrm);
            nrm = fmaf(bq.y, bq.y, nrm);
#ifdef USE_WMMA_F32
            v2f A; A[0] = aq.x; A[1] = aq.y;
            v2f B; B[0] = bq.x; B[1] = bq.y;
            acc = __builtin_amdgcn_wmma_f32_16x16x4_f32(
                /*neg_a=*/false, A, /*neg_b=*/false, B,
                /*c_mod=*/(short)0, acc, /*reuse_a=*/false, /*reuse_b=*/false);
#else
            dotp = fmaf(bq.x, aq.x, dotp);
            dotp = fmaf(bq.y, aq.y, dotp);
#endif
        }
        // each lane holds half of its key's norm; lane l and l^16 share a key
        nrm += __shfl_xor(nrm, 16, 32);
#ifdef USE_WMMA_F32
        const float dot = acc[0];               // D replicated over M rows
#else
        const float dot = dotp + __shfl_xor(dotp, 16, 32);
#endif
        if (lane < 16) {
            s = 2.0f * dot - nrm;
            sarr[tile * 16 + lane] = s;
        }
    }
    // wave max (lanes >=16 carry -inf, so no double counting)
#pragma unroll
    for (int off = 16; off; off >>= 1) s = fmaxf(s, __shfl_xor(s, off, 32));
    if (lane == 0) red[wave] = s;
    __syncthreads();
    if (threadIdx.x == 0) {
        float m = red[0];
#pragma unroll
        for (int w = 1; w < 8; ++w) m = fmaxf(m, red[w]);
        maxpart[blockIdx.x] = m;
    }
}

// ---------------- K2: global max reduce ----------------
__global__ void max_reduce_kernel(const float* __restrict__ maxpart, int n,
                                  float* __restrict__ smax) {
    __shared__ float red[256];
    float m = -INFINITY;
    for (int i = threadIdx.x; i < n; i += 256) m = fmaxf(m, maxpart[i]);
    red[threadIdx.x] = m;
    __syncthreads();
#pragma unroll
    for (int off = 128; off; off >>= 1) {
        if ((int)threadIdx.x < off)
            red[threadIdx.x] = fmaxf(red[threadIdx.x], red[threadIdx.x + off]);
        __syncthreads();
    }
    if (threadIdx.x == 0) *smax = red[0];
}

// ---------------- K3: streaming weighted accumulation over vals ----------------
// Block b owns rows [b*CHUNK, ...); 256 threads hold all 512 columns in regs.
__global__ void accum_kernel(const float* __restrict__ vals,
                             const float* __restrict__ sarr,
                             const float* __restrict__ smax_p,
                             float* __restrict__ mpart,
                             float* __restrict__ sumpart) {
    const float smax = *smax_p;
    const int t  = threadIdx.x;          // 256
    const int b  = blockIdx.x;           // 256 blocks
    const int j0 = b * CHUNK;
    int j1 = j0 + CHUNK; if (j1 > NMEM) j1 = NMEM;

    float a0 = 0.0f, a1 = 0.0f, se = 0.0f;
    for (int j = j0; j < j1; ++j) {
        const float e = __expf(sarr[j] - smax);
        const float* vp = vals + (size_t)j * H;
        __builtin_prefetch(vp + 8 * H + t, 0, 0);   // look ahead 8 rows
        a0 = fmaf(e, vp[t], a0);
        a1 = fmaf(e, vp[t + 256], a1);
        se += e;
    }
    mpart[(size_t)b * H + t]       = a0;
    mpart[(size_t)b * H + t + 256] = a1;
    if (t == 0) sumpart[b] = se;
}

// ---------------- K4: deterministic partial reduction + normalize ----------------
__global__ void finalize_m_kernel(const float* __restrict__ mpart,
                                  const float* __restrict__ sumpart,
                                  float* __restrict__ m_t) {
    const int h = threadIdx.x;           // 512
    float acc = 0.0f;
    for (int b = 0; b < RBLK; ++b) acc += mpart[(size_t)b * H + h];
    __shared__ float ssum;
    if (h == 0) {
        float s = 0.0f;
        for (int b = 0; b < RBLK; ++b) s += sumpart[b];
        ssum = s;
    }
    __syncthreads();
    m_t[h] = acc / ssum;
}

// ---------------- K5: LSTM gate GEMV (wave per output row) ----------------
__global__ void gates_kernel(const float* __restrict__ W_ih,
                             const float* __restrict__ W_hh,
                             const float* __restrict__ b_ih,
                             const float* __restrict__ b_hh,
                             const float* __restrict__ x_t,
                             const float* __restrict__ h0,
                             float* __restrict__ gates) {
    const int lane = threadIdx.x & 31;
    const int wave = threadIdx.x >> 5;
    const int o = blockIdx.x * 8 + wave;        // 2560 outputs / 320 blocks
    if (o >= NGATE) return;
    float acc = 0.0f;
    const float* wi = W_ih + (size_t)o * XDIM;
    for (int k = lane; k < XDIM; k += 32) acc = fmaf(wi[k], x_t[k], acc);
    const float* wh = W_hh + (size_t)o * H;
    for (int k = lane; k < H; k += 32) acc = fmaf(wh[k], h0[k], acc);
#pragma unroll
    for (int off = 16; off; off >>= 1) acc += __shfl_xor(acc, off, 32);
    if (lane == 0) gates[o] = acc + b_ih[o] + b_hh[o];
}

// ---------------- K6: LSTM elementwise + actor/critic heads ----------------
__global__ void final_kernel(const float* __restrict__ gates,
                             const float* __restrict__ c0,
                             const float* __restrict__ m_t,
                             const float* __restrict__ Wa, const float* __restrict__ ba,
                             const float* __restrict__ Wc, const float* __restrict__ bc,
                             float* __restrict__ out) {
    __shared__ float hs[H];
    const int t = threadIdx.x;               // 512
    const float ig = gates[t];
    const float fg = gates[H + t];
    const float gg = gates[2 * H + t];
    const float og = gates[3 * H + t];
    const float rg = gates[4 * H + t];
    const float c = sigmoidf_(fg) * c0[t]
                  + sigmoidf_(ig) * tanhf(gg)
                  + sigmoidf_(rg) * m_t[t];
    const float h = sigmoidf_(og) * tanhf(c);
    hs[t] = h;
    out[4 + t]   = h;     // h_t
    out[516 + t] = c;     // c_t
    __syncthreads();

    const int wave = t >> 5, lane = t & 31;
    if (wave < 4) {       // waves 0..2: action logits, wave 3: value
        const float* wrow = (wave < 3) ? (Wa + wave * H) : Wc;
        float acc = 0.0f;
        for (int k = lane; k < H; k += 32) acc = fmaf(wrow[k], hs[k], acc);
#pragma unroll
        for (int off = 16; off; off >>= 1) acc += __shfl_xor(acc, off, 32);
        if (lane == 0) {
            if (wave < 3) out[wave] = acc + ba[wave];
            else          out[3]    = acc + bc[0];
        }
    }
}

// ---------------- launcher ----------------
extern "C" void kernel_launch(void* const* d_in, const int* in_sizes, int n_in,
                              void* d_out, int out_size, void* d_ws, size_t ws_size,
                              hipStream_t stream) {
    (void)in_sizes; (void)n_in; (void)out_size; (void)ws_size;
    const float* obs  = (const float*)d_in[0];
    const float* pa   = (const float*)d_in[1];
    const float* pr   = (const float*)d_in[2];
    const float* h0   = (const float*)d_in[3];
    const float* c0   = (const float*)d_in[4];
    const float* keys = (const float*)d_in[5];
    const float* vals = (const float*)d_in[6];
    const float* W1   = (const float*)d_in[7];
    const float* b1   = (const float*)d_in[8];
    const float* W2   = (const float*)d_in[9];
    const float* b2   = (const float*)d_in[10];
    const float* W_ih = (const float*)d_in[11];
    const float* W_hh = (const float*)d_in[12];
    const float* b_ih = (const float*)d_in[13];
    const float* b_hh = (const float*)d_in[14];
    const float* Wa   = (const float*)d_in[15];
    const float* ba   = (const float*)d_in[16];
    const float* Wc   = (const float*)d_in[17];
    const float* bc   = (const float*)d_in[18];

    float* ws  = (float*)d_ws;
    float* out = (float*)d_out;

    encoder_kernel<<<1, 64, 0, stream>>>(obs, pa, pr, W1, b1, W2, b2,
                                         ws + OFF_FEATS, ws + OFF_XT, out);
    score_kernel<<<SCORE_BLOCKS, 256, 0, stream>>>(keys, ws + OFF_FEATS,
                                                   ws + OFF_S, ws + OFF_MAXP);
    max_reduce_kernel<<<1, 256, 0, stream>>>(ws + OFF_MAXP, SCORE_BLOCKS,
                                             ws + OFF_SMAX);
    accum_kernel<<<RBLK, 256, 0, stream>>>(vals, ws + OFF_S, ws + OFF_SMAX,
                                           ws + OFF_MPART, ws + OFF_SUMP);
    finalize_m_kernel<<<1, H, 0, stream>>>(ws + OFF_MPART, ws + OFF_SUMP,
                                           ws + OFF_MT);
    gates_kernel<<<NGATE / 8, 256, 0, stream>>>(W_ih, W_hh, b_ih, b_hh,
                                                ws + OFF_XT, h0, ws + OFF_GATES);
    final_kernel<<<1, H, 0, stream>>>(ws + OFF_GATES, c0, ws + OFF_MT,
                                      Wa, ba, Wc, bc, out);
}